// SURVEY_BA_2grid_GCN_21930103013658
// MI455X (gfx1250) — compile-verified
//
#include <hip/hip_runtime.h>
#include <hip/hip_bf16.h>
#include <math.h>

#define HF 30      // logical hidden feature width
#define LD 32      // padded feature leading dimension (power of two)
#define F_IN 10    // input feature width
#define KP1 12     // padded K for layer 1 (F_IN -> multiple of 4)

typedef float v2f __attribute__((ext_vector_type(2)));
typedef float v8f __attribute__((ext_vector_type(8)));

// ---------------------------------------------------------------- elementwise
__global__ void k_fill(float* __restrict__ p, float v, int n) {
    int i = blockIdx.x * blockDim.x + threadIdx.x;
    if (i < n) p[i] = v;
}

__global__ void k_fill4(float4* __restrict__ p, float v, int n4) {
    int i = blockIdx.x * blockDim.x + threadIdx.x;
    if (i < n4) p[i] = make_float4(v, v, v, v);
}

__global__ void k_deg_edges(const int* __restrict__ colE, const float* __restrict__ w,
                            float* __restrict__ deg, int E) {
    int e = blockIdx.x * blockDim.x + threadIdx.x;
    if (e < E)
        __hip_atomic_fetch_add(&deg[colE[e]], w[e], __ATOMIC_RELAXED, __HIP_MEMORY_SCOPE_AGENT);
}

__global__ void k_dis(const float* __restrict__ deg, float* __restrict__ dis, int n) {
    int i = blockIdx.x * blockDim.x + threadIdx.x;
    if (i < n) {
        float d = deg[i];
        dis[i] = d > 0.0f ? rsqrtf(d) : 0.0f;
    }
}

// pad x (N x F_IN) -> xpad (N x KP1), zero-filled tail columns
__global__ void k_pad_x(const float* __restrict__ x, float* __restrict__ xpad, int total) {
    int idx = blockIdx.x * blockDim.x + threadIdx.x;
    if (idx >= total) return;
    int node = idx / KP1;
    int k    = idx - node * KP1;
    xpad[idx] = (k < F_IN) ? x[node * F_IN + k] : 0.0f;
}

// pad W (K x HF) -> Wpad (KP x LD), zero-filled; zero B columns make the GEMM
// write exact zeros into pad columns of the output, keeping padding invariant.
__global__ void k_pad_w(const float* __restrict__ W, float* __restrict__ Wp,
                        int K, int total) {
    int idx = blockIdx.x * blockDim.x + threadIdx.x;
    if (idx >= total) return;
    int k = idx >> 5;        // LD == 32
    int c = idx & (LD - 1);
    Wp[idx] = (k < K && c < HF) ? W[k * HF + c] : 0.0f;
}

// ------------------------------------------------- dense transform, WMMA f32
// A: n x KP row-major (zero-padded), Wp: KP x LD (zero-padded),
// T: n x LD row-major. One wave per 16x16 output tile, KP consumed 4/step by
// V_WMMA_F32_16X16X4_F32, fully unrolled, no per-element guards.
template <int KP>
__global__ void k_gemm_wmma(const float* __restrict__ A, const float* __restrict__ Wp,
                            float* __restrict__ T, int n) {
    int wid  = (blockIdx.x * blockDim.x + threadIdx.x) >> 5;  // wave-uniform
    int lane = threadIdx.x & 31;
    int tiles_m = (n + 15) >> 4;
    int ntask   = tiles_m * 2;           // two 16-wide column tiles cover LD=32
    if (wid >= ntask) return;            // wave-uniform exit: EXEC stays all-1s

    int tm = wid >> 1;
    int cb = (wid & 1) << 4;             // output column base: 0 or 16
    int m0 = tm << 4;

    int l15   = lane & 15;
    int khalf = (lane >> 4) << 1;        // 0 for lanes 0-15, 2 for lanes 16-31
    int mrow  = m0 + l15;                // A-fragment row for this lane
    if (mrow >= n) mrow = n - 1;         // clamp (stores are guarded below)
    int ncol  = cb + l15;                // B-fragment column for this lane

    const v2f* __restrict__ Arow = (const v2f*)(A + (size_t)mrow * KP);

    v8f acc = {};
    #pragma unroll
    for (int kb = 0; kb < KP; kb += 4) {
        int k0 = kb + khalf;
        v2f a = Arow[k0 >> 1];                      // A[m][k0], A[m][k0+1]
        v2f b;
        b.x = Wp[(k0    ) * LD + ncol];             // B row k0
        b.y = Wp[(k0 + 1) * LD + ncol];             // B row k0+1
        acc = __builtin_amdgcn_wmma_f32_16x16x4_f32(
            /*neg_a=*/false, a, /*neg_b=*/false, b,
            /*c_mod=*/(short)0, acc, /*reuse_a=*/false, /*reuse_b=*/false);
    }

    // C/D 16x16: VGPR v -> row m0+v (lanes 0-15) or m0+8+v (lanes 16-31)
    int mbase = m0 + ((lane >> 4) << 3);
    if (m0 + 16 <= n) {
        // wave-uniform fast path: 8 unguarded stores, one base + imm offsets
        float* __restrict__ p = T + (size_t)mbase * LD + ncol;
        #pragma unroll
        for (int v = 0; v < 8; ++v) p[(size_t)v * LD] = acc[v];
    } else {
        #pragma unroll
        for (int v = 0; v < 8; ++v) {
            int m = mbase + v;
            if (m < n) T[(size_t)m * LD + ncol] = acc[v];
        }
    }
}

// -------------------------------------------------- edge scatter
// One wave per edge (lanes = features, ld=32). Edge metadata goes through
// readfirstlane -> wave-uniform scalar loads; gather + fp32 atomic stay vector.
__global__ void k_scatter(const int* __restrict__ rowE, const int* __restrict__ colE,
                          const float* __restrict__ w, const float* __restrict__ dis,
                          const float* __restrict__ T, float* __restrict__ agg,
                          int E) {
    long long idx = (long long)blockIdx.x * blockDim.x + threadIdx.x;
    int f = (int)(idx & (LD - 1));
    int e = (int)(idx >> 5);
    if (e >= E || f >= HF) return;
    int eu = __builtin_amdgcn_readfirstlane(e);     // uniform across the wave
    int r = rowE[eu], c = colE[eu];
    float coeff = dis[r] * w[eu] * dis[c];
    __hip_atomic_fetch_add(&agg[(size_t)c * LD + f], coeff * T[(size_t)r * LD + f],
                           __ATOMIC_RELAXED, __HIP_MEMORY_SCOPE_AGENT);
}

// self-loop contribution + bias + ReLU epilogue; pad columns come out as 0
__global__ void k_finish(const float* __restrict__ agg, const float* __restrict__ T,
                         const float* __restrict__ dis, const float* __restrict__ b,
                         float* __restrict__ hout, int total) {
    int idx = blockIdx.x * blockDim.x + threadIdx.x;
    if (idx >= total) return;
    int node = idx >> 5;
    int f    = idx & (LD - 1);
    float s  = dis[node];
    float bv = (f < HF) ? b[f] : 0.0f;
    float v  = agg[idx] + s * s * T[idx] + bv;
    hout[idx] = fmaxf(v, 0.0f);
}

// ---------------------------------------------------------------- pooling
// Post-ReLU values are >= 0, so signed-int atomicMax on the float bit
// pattern with 0.0f init reproduces segment_max for non-empty segments.
__global__ void k_pool(const float* __restrict__ hfeat, const int* __restrict__ batch,
                       int* __restrict__ pooled, int total) {
    int idx = blockIdx.x * blockDim.x + threadIdx.x;
    if (idx >= total) return;
    int node = idx >> 5;
    int f    = idx & (LD - 1);
    if (f >= HF) return;
    int g = batch[node];
    atomicMax(&pooled[g * HF + f], __float_as_int(hfeat[idx]));
}

// ---------------------------------------------------------------- MLP head
__global__ void k_head(const float* __restrict__ pooled,
                       const float* __restrict__ LW1, const float* __restrict__ Lb1,
                       const float* __restrict__ LW2, const float* __restrict__ Lb2,
                       float* __restrict__ out, int G) {
    int g = blockIdx.x * blockDim.x + threadIdx.x;
    if (g >= G) return;
    float p[HF];
    #pragma unroll
    for (int i = 0; i < HF; ++i) p[i] = pooled[g * HF + i];
    float z[10];
    #pragma unroll
    for (int j = 0; j < 10; ++j) {
        float acc = Lb1[j];
        #pragma unroll
        for (int i = 0; i < HF; ++i) acc += p[i] * LW1[i * 10 + j];
        z[j] = acc > 0.0f ? acc : 0.0f;
    }
    #pragma unroll
    for (int c = 0; c < 2; ++c) {
        float acc = Lb2[c];
        #pragma unroll
        for (int j = 0; j < 10; ++j) acc += z[j] * LW2[j * 2 + c];
        out[g * 2 + c] = acc;
    }
}

// ---------------------------------------------------------------- launcher
extern "C" void kernel_launch(void* const* d_in, const int* in_sizes, int n_in,
                              void* d_out, int out_size, void* d_ws, size_t ws_size,
                              hipStream_t stream) {
    const float* x     = (const float*)d_in[0];
    const int*   ei    = (const int*)  d_in[1];   // [2, E]: row then col
    const int*   batch = (const int*)  d_in[2];
    const float* ew    = (const float*)d_in[3];
    const float* W1    = (const float*)d_in[4];
    const float* b1    = (const float*)d_in[5];
    const float* W2    = (const float*)d_in[6];
    const float* b2    = (const float*)d_in[7];
    const float* W3    = (const float*)d_in[8];
    const float* b3    = (const float*)d_in[9];
    const float* LW1   = (const float*)d_in[10];
    const float* Lb1   = (const float*)d_in[11];
    const float* LW2   = (const float*)d_in[12];
    const float* Lb2   = (const float*)d_in[13];
    float* out = (float*)d_out;

    const int N = in_sizes[0] / F_IN;
    const int E = in_sizes[3];
    const int G = out_size / 2;
    const int* rowE = ei;
    const int* colE = ei + E;

    float* ws     = (float*)d_ws;
    float* deg    = ws;
    float* dis    = deg  + N;
    float* xpad   = dis  + N;                   // N * KP1
    float* t      = xpad + (size_t)N * KP1;     // N * LD
    float* agg    = t    + (size_t)N * LD;      // N * LD
    float* hbuf   = agg  + (size_t)N * LD;      // N * LD
    float* Wpad   = hbuf + (size_t)N * LD;      // LD * LD
    float* pooled = Wpad + LD * LD;             // G * HF

    const dim3 blk(256);
    auto nb = [](long long n) { return dim3((unsigned)((n + 255) / 256)); };

    const long long NL = (long long)N * LD;
    const long long EL = (long long)E * LD;

    // symmetric normalization coefficients (self-loop weight 1 folded into init)
    k_fill     <<<nb(N), blk, 0, stream>>>(deg, 1.0f, N);
    k_deg_edges<<<nb(E), blk, 0, stream>>>(colE, ew, deg, E);
    k_dis      <<<nb(N), blk, 0, stream>>>(deg, dis, N);

    // pad x once
    k_pad_x<<<nb((long long)N * KP1), blk, 0, stream>>>(x, xpad, N * KP1);

    const float* Ws[3] = {W1, W2, W3};
    const float* bs[3] = {b1, b2, b3};
    const int    Ks[3] = {F_IN, HF, HF};
    const float* hin = xpad;

    const long long gemmThreads = (long long)(((N + 15) / 16) * 2) * 32;
    for (int l = 0; l < 3; ++l) {
        const int KPl = (l == 0) ? KP1 : LD;
        k_pad_w<<<nb(KPl * LD), blk, 0, stream>>>(Ws[l], Wpad, Ks[l], KPl * LD);
        if (l == 0)
            k_gemm_wmma<KP1><<<nb(gemmThreads), blk, 0, stream>>>(hin, Wpad, t, N);
        else
            k_gemm_wmma<LD><<<nb(gemmThreads), blk, 0, stream>>>(hin, Wpad, t, N);
        k_fill4  <<<nb(NL / 4), blk, 0, stream>>>((float4*)agg, 0.0f, (int)(NL / 4));
        k_scatter<<<nb(EL), blk, 0, stream>>>(rowE, colE, ew, dis, t, agg, E);
        k_finish <<<nb(NL), blk, 0, stream>>>(agg, t, dis, bs[l], hbuf, (int)NL);
        hin = hbuf;
    }

    // max pooling per graph, then 30->10->2 head
    k_fill<<<nb((long long)G * HF), blk, 0, stream>>>(pooled, 0.0f, G * HF);
    k_pool<<<nb(NL), blk, 0, stream>>>(hbuf, batch, (int*)pooled, (int)NL);
    k_head<<<nb(G), blk, 0, stream>>>(pooled, LW1, Lb1, LW2, Lb2, out, G);
}